// MemoryAccess_32684701123315
// MI455X (gfx1250) — compile-verified
//
#include <hip/hip_runtime.h>

typedef float v2f __attribute__((ext_vector_type(2)));
typedef float v8f __attribute__((ext_vector_type(8)));
typedef int   v4i __attribute__((ext_vector_type(4)));

#define BATCH 64
#define SLOT 512
#define AVAIL 10000
#define RS 3
#define KCHUNK 64
#define NTILE 32

// ---------------------------------------------------------------------------
// CDNA5 async global->LDS copy (b128) with sync fallback, + asynccnt waits.
// Builtin signature (from hipcc diagnostic): param0 = int4 AS1 (__device__),
// param1 = int4 AS3 (__shared__), then imm offset, imm cpol.
// ---------------------------------------------------------------------------
#if __has_builtin(__builtin_amdgcn_global_load_async_to_lds_b128)
#define HAVE_ASYNC_LDS 1
#else
#define HAVE_ASYNC_LDS 0
#endif

__device__ __forceinline__ void async_copy_b128(const float* src, float* ldsDst) {
#if HAVE_ASYNC_LDS
  __builtin_amdgcn_global_load_async_to_lds_b128(
      (__attribute__((address_space(1))) v4i*)src,
      (__attribute__((address_space(3))) v4i*)ldsDst, 0, 0);
#else
  *(float4*)ldsDst = *(const float4*)src;
#endif
}

__device__ __forceinline__ void wait_async_le2() {
#if HAVE_ASYNC_LDS
#if __has_builtin(__builtin_amdgcn_s_wait_asynccnt)
  __builtin_amdgcn_s_wait_asynccnt(2);
#else
  asm volatile("s_wait_asynccnt 0x2" ::: "memory");
#endif
#endif
}

__device__ __forceinline__ void wait_async_0() {
#if HAVE_ASYNC_LDS
#if __has_builtin(__builtin_amdgcn_s_wait_asynccnt)
  __builtin_amdgcn_s_wait_asynccnt(0);
#else
  asm volatile("s_wait_asynccnt 0x0" ::: "memory");
#endif
#endif
}

// ---------------------------------------------------------------------------
// fa "pre" stage: t1 = x@W1+b1 (64x16), a = softmax(t1), g = a@W2+b2 (64xd),
// h = [x, x*g] (64 x 2d).  Tiny (<= 2 MFLOP) -> one workgroup.
// ---------------------------------------------------------------------------
__global__ __launch_bounds__(256) void fa_pre_kernel(
    const float* __restrict__ x, const float* __restrict__ W1,
    const float* __restrict__ b1, const float* __restrict__ W2,
    const float* __restrict__ b2, float* __restrict__ h, int d) {
  __shared__ float t1[BATCH * 16];
  __shared__ float att[BATCH * 16];
  for (int i = threadIdx.x; i < BATCH * 16; i += blockDim.x) {
    int bi = i >> 4, f = i & 15;
    float s = b1[f];
    const float* xr = x + bi * d;
    for (int k = 0; k < d; ++k) s += xr[k] * W1[k * 16 + f];
    t1[i] = s;
  }
  __syncthreads();
  if (threadIdx.x < BATCH) {
    int bi = threadIdx.x;
    float mx = -3.4e38f;
    for (int f = 0; f < 16; ++f) mx = fmaxf(mx, t1[bi * 16 + f]);
    float s = 0.f;
    for (int f = 0; f < 16; ++f) { float e = __expf(t1[bi * 16 + f] - mx); att[bi * 16 + f] = e; s += e; }
    float inv = 1.f / s;
    for (int f = 0; f < 16; ++f) att[bi * 16 + f] *= inv;
  }
  __syncthreads();
  for (int i = threadIdx.x; i < BATCH * d; i += blockDim.x) {
    int bi = i / d, j = i - bi * d;
    float g = b2[j];
    for (int f = 0; f < 16; ++f) g += att[bi * 16 + f] * W2[f * d + j];
    float xv = x[i];
    h[bi * 2 * d + j] = xv;
    h[bi * 2 * d + d + j] = xv * g;
  }
}

// ---------------------------------------------------------------------------
// WMMA GEMM: C(64 x N) = act(A(64 x K) @ W(K x N) + bias), f32 via
// V_WMMA_F32_16X16X4_F32.  256 threads = 8 waves; wave -> (M-tile, N-subtile);
// W streamed HBM -> LDS with double-buffered GLOBAL_LOAD_ASYNC_TO_LDS_B128
// (2 issues/thread/chunk -> exact ASYNCcnt accounting), WMMA consumes the
// other buffer.  Requires K % 64 == 0; N % 16 == 0 (OOB 16-col subtiles in
// the last block are clamped on load, skipped on store; EXEC stays all-1s
// around every WMMA).  act: 0 = none, 1 = relu.
// ---------------------------------------------------------------------------
__global__ __launch_bounds__(256) void gemm_wmma_kernel(
    const float* __restrict__ A, const float* __restrict__ W,
    const float* __restrict__ bias, float* __restrict__ C,
    int K, int N, int act) {
  __shared__ float Bs[2][KCHUNK * NTILE];  // 2 x 8 KB
  const int lane = threadIdx.x & 31;
  const int wave = threadIdx.x >> 5;
  const int mt = wave & 3, nt = wave >> 2;
  const int n0 = blockIdx.x * NTILE;
  const int m0 = mt * 16;
  const int row = m0 + (lane & 15);
  const int khalf = (lane >> 4) * 2;  // 0 or 2

  // stage W[k0:k0+64, n0:n0+32] into Bs[buf]: 512 b128 ops, 2 per thread
  auto stage = [&](int k0, int buf) {
#pragma unroll
    for (int t = 0; t < 2; ++t) {
      int idx = threadIdx.x + t * 256;        // 0..511
      int kk = idx >> 3, c4 = (idx & 7) * 4;  // chunk row, col (float4 units)
      int col = n0 + c4;
      if (col >= N) col = 0;  // fully-OOB subtile -> harmless in-bounds read
      async_copy_b128(W + (size_t)(k0 + kk) * N + col, &Bs[buf][kk * NTILE + c4]);
    }
  };

  stage(0, 0);
  v8f acc = {};
  const int nsteps = K / KCHUNK;
  for (int i = 0; i < nsteps; ++i) {
    const int k0 = i * KCHUNK;
    if (i + 1 < nsteps) {
      stage(k0 + KCHUNK, (i + 1) & 1);
      wait_async_le2();  // chunk i resident (in-order completion)
    } else {
      wait_async_0();
    }
    __syncthreads();
    const float* ap = A + (size_t)row * K + k0 + khalf;
    const float* bp = &Bs[i & 1][nt * 16 + (lane & 15)];
#pragma unroll 4
    for (int kk = 0; kk < KCHUNK; kk += 4) {
      v2f a;
      a.x = ap[kk + 0];
      a.y = ap[kk + 1];
      v2f b;
      b.x = bp[(kk + khalf + 0) * NTILE];
      b.y = bp[(kk + khalf + 1) * NTILE];
      acc = __builtin_amdgcn_wmma_f32_16x16x4_f32(false, a, false, b,
                                                  (short)0, acc, false, false);
    }
    __syncthreads();  // all waves done with Bs[i&1] before it is re-staged
  }
  const int n = n0 + nt * 16 + (lane & 15);
  if (n < N) {
    const float bv = bias[n];
    const int mbase = m0 + ((lane >> 4) ? 8 : 0);
#pragma unroll
    for (int r = 0; r < 8; ++r) {
      float v = acc[r] + bv;
      if (act == 1) v = fmaxf(v, 0.f);
      C[(size_t)(mbase + r) * N + n] = v;
    }
  }
}

// ---------------------------------------------------------------------------
// LayerNorm over last dim: z = (y-mu)/sqrt(var+1e-5)*g + b.  1 block / row.
// ---------------------------------------------------------------------------
__global__ __launch_bounds__(256) void ln_kernel(
    const float* __restrict__ y, const float* __restrict__ g,
    const float* __restrict__ b, float* __restrict__ z, int d) {
  __shared__ float red[256];
  const int row = blockIdx.x;
  const float* yr = y + (size_t)row * d;
  float s = 0.f;
  for (int j = threadIdx.x; j < d; j += 256) s += yr[j];
  red[threadIdx.x] = s; __syncthreads();
  for (int st = 128; st > 0; st >>= 1) {
    if (threadIdx.x < st) red[threadIdx.x] += red[threadIdx.x + st];
    __syncthreads();
  }
  const float mu = red[0] / d;
  __syncthreads();
  float v = 0.f;
  for (int j = threadIdx.x; j < d; j += 256) { float t = yr[j] - mu; v += t * t; }
  red[threadIdx.x] = v; __syncthreads();
  for (int st = 128; st > 0; st >>= 1) {
    if (threadIdx.x < st) red[threadIdx.x] += red[threadIdx.x + st];
    __syncthreads();
  }
  const float inv = rsqrtf(red[0] / d + 1e-5f);
  for (int j = threadIdx.x; j < d; j += 256)
    z[(size_t)row * d + j] = (yr[j] - mu) * inv * g[j] + b[j];
}

// Small projection (N=3): out = act(A(64xK)@W(KxN)+b), act 2 = sigmoid.
__global__ void small_gemm_kernel(const float* __restrict__ A,
                                  const float* __restrict__ W,
                                  const float* __restrict__ bias,
                                  float* __restrict__ C, int K, int N, int act) {
  int i = blockIdx.x * blockDim.x + threadIdx.x;
  if (i >= BATCH * N) return;
  int bi = i / N, n = i - bi * N;
  float s = bias[n];
  const float* ar = A + (size_t)bi * K;
  for (int k = 0; k < K; ++k) s += ar[k] * W[k * N + n];
  if (act == 1) s = fmaxf(s, 0.f);
  else if (act == 2) s = 1.f / (1.f + __expf(-s));
  C[i] = s;
}

// ---------------------------------------------------------------------------
// Read-head double softmax.  Stage1: per (row,chunk) max/argmax/sumexp.
// Stage2: per row combine -> global softmax denom.  Stage3: per (row,chunk)
// second softmax denom in p-space -> rw.
// ---------------------------------------------------------------------------
__global__ __launch_bounds__(256) void read_stage1_kernel(
    const float* __restrict__ scores, float* __restrict__ cmax,
    int* __restrict__ carg, float* __restrict__ csum) {
  __shared__ float smx[256]; __shared__ int sid[256]; __shared__ float ssm[256];
  const int row = blockIdx.y, c = blockIdx.x;
  const float* p = scores + (size_t)row * (RS * AVAIL) + (size_t)c * AVAIL;
  float mx = -3.4e38f; int arg = 0;
  for (int j = threadIdx.x; j < AVAIL; j += 256) {
    float v = p[j];
    if (v > mx) { mx = v; arg = j; }
  }
  smx[threadIdx.x] = mx; sid[threadIdx.x] = arg; __syncthreads();
  for (int st = 128; st > 0; st >>= 1) {
    if (threadIdx.x < st) {
      float v = smx[threadIdx.x + st]; int a = sid[threadIdx.x + st];
      if (v > smx[threadIdx.x] ||
          (v == smx[threadIdx.x] && a < sid[threadIdx.x])) {
        smx[threadIdx.x] = v; sid[threadIdx.x] = a;
      }
    }
    __syncthreads();
  }
  const float M = smx[0];
  float s = 0.f;
  for (int j = threadIdx.x; j < AVAIL; j += 256) s += __expf(p[j] - M);
  ssm[threadIdx.x] = s; __syncthreads();
  for (int st = 128; st > 0; st >>= 1) {
    if (threadIdx.x < st) ssm[threadIdx.x] += ssm[threadIdx.x + st];
    __syncthreads();
  }
  if (threadIdx.x == 0) {
    cmax[row * RS + c] = M; carg[row * RS + c] = sid[0]; csum[row * RS + c] = ssm[0];
  }
}

__global__ void read_stage2_kernel(const float* __restrict__ cmax,
                                   const float* __restrict__ csum,
                                   float* __restrict__ rowM, float* __restrict__ rowS) {
  int row = threadIdx.x;
  if (row >= BATCH) return;
  float M = fmaxf(cmax[row * RS], fmaxf(cmax[row * RS + 1], cmax[row * RS + 2]));
  float S = 0.f;
  for (int c = 0; c < RS; ++c) S += __expf(cmax[row * RS + c] - M) * csum[row * RS + c];
  rowM[row] = M; rowS[row] = S;
}

__global__ __launch_bounds__(256) void read_stage3_kernel(
    const float* __restrict__ scores, const float* __restrict__ cmax,
    const float* __restrict__ rowM, const float* __restrict__ rowS,
    float* __restrict__ rw) {
  __shared__ float ssm[256];
  const int row = blockIdx.y, c = blockIdx.x;
  const float* p = scores + (size_t)row * (RS * AVAIL) + (size_t)c * AVAIL;
  const float M = rowM[row], invS = 1.f / rowS[row];
  float s = 0.f;
  for (int j = threadIdx.x; j < AVAIL; j += 256)
    s += __expf(__expf(p[j] - M) * invS);
  ssm[threadIdx.x] = s; __syncthreads();
  for (int st = 128; st > 0; st >>= 1) {
    if (threadIdx.x < st) ssm[threadIdx.x] += ssm[threadIdx.x + st];
    __syncthreads();
  }
  if (threadIdx.x == 0) {
    float pmax = __expf(cmax[row * RS + c] - M) * invS;
    rw[row * RS + c] = __expf(pmax) / ssm[0];
  }
}

// ---------------------------------------------------------------------------
// Elementwise glue kernels
// ---------------------------------------------------------------------------
__global__ void zero_kernel(float* p, int n) {
  int i = blockIdx.x * blockDim.x + threadIdx.x;
  if (i < n) p[i] = 0.f;
}

// x_out(64x1024) = [memory[idx[b][s]], inputs[b]]
__global__ void concat_gather_kernel(const float* __restrict__ memory,
                                     const int* __restrict__ idx, int s,
                                     const float* __restrict__ inputs,
                                     float* __restrict__ xout) {
  int i = blockIdx.x * blockDim.x + threadIdx.x;
  if (i >= BATCH * 2 * SLOT) return;
  int b = i >> 10, j = i & 1023;
  float v = (j < SLOT) ? memory[(size_t)idx[b * RS + s] * SLOT + j]
                       : inputs[(size_t)b * SLOT + (j - SLOT)];
  xout[i] = v;
}

// x_out(64x1024) = [r[b], m[b]]
__global__ void concat2_kernel(const float* __restrict__ r,
                               const float* __restrict__ m,
                               float* __restrict__ xout) {
  int i = blockIdx.x * blockDim.x + threadIdx.x;
  if (i >= BATCH * 2 * SLOT) return;
  int b = i >> 10, j = i & 1023;
  xout[i] = (j < SLOT) ? r[(size_t)b * SLOT + j] : m[(size_t)b * SLOT + (j - SLOT)];
}

// r = u*w + mem*(1-w), w = uw[b][s], mem = memory[idx[b][s]]
__global__ void mix_r_kernel(const float* __restrict__ u, const float* __restrict__ uw,
                             int s, const float* __restrict__ memory,
                             const int* __restrict__ idx, float* __restrict__ r) {
  int i = blockIdx.x * blockDim.x + threadIdx.x;
  if (i >= BATCH * SLOT) return;
  int b = i >> 9, j = i & 511;
  float w = uw[b * RS + s];
  float mv = memory[(size_t)idx[b * RS + s] * SLOT + j];
  r[i] = u[i] * w + mv * (1.f - w);
}

// m += t * rw[b][s]
__global__ void accum_m_kernel(const float* __restrict__ t, const float* __restrict__ rw,
                               int s, float* __restrict__ m) {
  int i = blockIdx.x * blockDim.x + threadIdx.x;
  if (i >= BATCH * SLOT) return;
  int b = i >> 9;
  m[i] += t[i] * rw[b * RS + s];
}

// ---------------------------------------------------------------------------
// Host-side orchestration
// ---------------------------------------------------------------------------
struct Head {
  const float *W1, *b1, *W2, *b2, *Wf, *bf, *ln_g, *ln_b, *W, *b;
  int d, dout;
};

static Head head_at(void* const* d_in, int hi, int d, int dout) {
  int base = 3 + hi * 10;
  Head h;
  h.W1 = (const float*)d_in[base + 0]; h.b1 = (const float*)d_in[base + 1];
  h.W2 = (const float*)d_in[base + 2]; h.b2 = (const float*)d_in[base + 3];
  h.Wf = (const float*)d_in[base + 4]; h.bf = (const float*)d_in[base + 5];
  h.ln_g = (const float*)d_in[base + 6]; h.ln_b = (const float*)d_in[base + 7];
  h.W = (const float*)d_in[base + 8]; h.b = (const float*)d_in[base + 9];
  h.d = d; h.dout = dout;
  return h;
}

static inline int gblocks(int N) { return (N + NTILE - 1) / NTILE; }

// pre -> gemm1(relu) -> layernorm; result in zbuf (64 x d)
static void run_fa(hipStream_t st, const Head& H, const float* x,
                   float* hbuf, float* ybuf, float* zbuf) {
  fa_pre_kernel<<<1, 256, 0, st>>>(x, H.W1, H.b1, H.W2, H.b2, hbuf, H.d);
  gemm_wmma_kernel<<<gblocks(H.d), 256, 0, st>>>(hbuf, H.Wf, H.bf, ybuf, 2 * H.d, H.d, 1);
  ln_kernel<<<BATCH, 256, 0, st>>>(ybuf, H.ln_g, H.ln_b, zbuf, H.d);
}

extern "C" void kernel_launch(void* const* d_in, const int* in_sizes, int n_in,
                              void* d_out, int out_size, void* d_ws, size_t ws_size,
                              hipStream_t stream) {
  (void)in_sizes; (void)n_in; (void)out_size; (void)ws_size;
  const float* inputs = (const float*)d_in[0];
  const float* memory = (const float*)d_in[1];
  const int* rand_idx = (const int*)d_in[2];

  Head read_w      = head_at(d_in, 0, 512, RS * AVAIL);
  Head upd_w       = head_at(d_in, 1, 512, RS);
  Head upd_w_rand  = head_at(d_in, 2, 512, RS);
  Head upd_mem     = head_at(d_in, 3, 1024, SLOT);
  Head upd_mem_r   = head_at(d_in, 4, 1024, SLOT);
  Head apply_mem   = head_at(d_in, 5, 1024, SLOT);
  Head apply_mem_r = head_at(d_in, 6, 1024, SLOT);

  float* ws = (float*)d_ws;
  size_t off = 0;
  auto alloc = [&](size_t n) { float* p = ws + off; off += n; return p; };
  float* scores = alloc((size_t)BATCH * RS * AVAIL);  // 64 x 30000
  float* hbuf   = alloc((size_t)BATCH * 2048);
  float* ybuf   = alloc((size_t)BATCH * 1024);
  float* zbuf   = alloc((size_t)BATCH * 1024);
  float* x1     = alloc((size_t)BATCH * 1024);
  float* ubuf   = alloc((size_t)BATCH * SLOT);
  float* rbuf   = alloc((size_t)BATCH * SLOT);
  float* tbuf   = alloc((size_t)BATCH * SLOT);
  float* cmax   = alloc(BATCH * RS);
  float* csum   = alloc(BATCH * RS);
  float* rowM   = alloc(BATCH);
  float* rowS   = alloc(BATCH);
  float* rw     = alloc(BATCH * RS);
  float* uw     = alloc(BATCH * RS);
  float* uwr    = alloc(BATCH * RS);
  int*   carg   = (int*)alloc(BATCH * RS);
  float* m      = (float*)d_out;  // m accumulates directly in d_out

  const int EW1 = (BATCH * 2 * SLOT + 255) / 256;  // 65536-elem grids
  const int EW2 = (BATCH * SLOT + 255) / 256;      // 32768-elem grids

  // ---- read head: scores = head(read_w, inputs), then double softmax ----
  run_fa(stream, read_w, inputs, hbuf, ybuf, zbuf);
  gemm_wmma_kernel<<<gblocks(RS * AVAIL), 256, 0, stream>>>(
      zbuf, read_w.W, read_w.b, scores, 512, RS * AVAIL, 0);
  read_stage1_kernel<<<dim3(RS, BATCH), 256, 0, stream>>>(scores, cmax, carg, csum);
  read_stage2_kernel<<<1, BATCH, 0, stream>>>(cmax, csum, rowM, rowS);
  read_stage3_kernel<<<dim3(RS, BATCH), 256, 0, stream>>>(scores, cmax, rowM, rowS, rw);

  // ---- gate heads: uw, uwr = sigmoid(head(., inputs)) ----
  run_fa(stream, upd_w, inputs, hbuf, ybuf, zbuf);
  small_gemm_kernel<<<1, 256, 0, stream>>>(zbuf, upd_w.W, upd_w.b, uw, 512, RS, 2);
  run_fa(stream, upd_w_rand, inputs, hbuf, ybuf, zbuf);
  small_gemm_kernel<<<1, 256, 0, stream>>>(zbuf, upd_w_rand.W, upd_w_rand.b, uwr, 512, RS, 2);

  zero_kernel<<<EW2, 256, 0, stream>>>(m, BATCH * SLOT);

  // ---- sequential slot loop ----
  for (int s = 0; s < RS; ++s) {
    // branch A (argmax-read slots)
    concat_gather_kernel<<<EW1, 256, 0, stream>>>(memory, carg, s, inputs, x1);
    run_fa(stream, upd_mem, x1, hbuf, ybuf, zbuf);
    gemm_wmma_kernel<<<gblocks(SLOT), 256, 0, stream>>>(zbuf, upd_mem.W, upd_mem.b, ubuf, 1024, SLOT, 1);
    mix_r_kernel<<<EW2, 256, 0, stream>>>(ubuf, uw, s, memory, carg, rbuf);
    concat2_kernel<<<EW1, 256, 0, stream>>>(rbuf, m, x1);
    run_fa(stream, apply_mem, x1, hbuf, ybuf, zbuf);
    gemm_wmma_kernel<<<gblocks(SLOT), 256, 0, stream>>>(zbuf, apply_mem.W, apply_mem.b, tbuf, 1024, SLOT, 1);
    accum_m_kernel<<<EW2, 256, 0, stream>>>(tbuf, rw, s, m);

    // branch B (random slots)
    concat_gather_kernel<<<EW1, 256, 0, stream>>>(memory, rand_idx, s, inputs, x1);
    run_fa(stream, upd_mem_r, x1, hbuf, ybuf, zbuf);
    gemm_wmma_kernel<<<gblocks(SLOT), 256, 0, stream>>>(zbuf, upd_mem_r.W, upd_mem_r.b, ubuf, 1024, SLOT, 1);
    mix_r_kernel<<<EW2, 256, 0, stream>>>(ubuf, uwr, s, memory, rand_idx, rbuf);
    concat2_kernel<<<EW1, 256, 0, stream>>>(rbuf, m, x1);
    run_fa(stream, apply_mem_r, x1, hbuf, ybuf, zbuf);
    gemm_wmma_kernel<<<gblocks(SLOT), 256, 0, stream>>>(zbuf, apply_mem_r.W, apply_mem_r.b, tbuf, 1024, SLOT, 1);
    accum_m_kernel<<<EW2, 256, 0, stream>>>(tbuf, rw, s, m);
  }
}